// MultiScaleDeformableAttention_1_74423193305551
// MI455X (gfx1250) — compile-verified
//
#include <hip/hip_runtime.h>
#include <hip/hip_bf16.h>

// MSDA for MI455X (gfx1250, wave32).
// - 1 wave per (b, q, h). lane = channel for gathers (coalesced 64B half-rows).
// - Bilinear+attention weighted reduction over 16 points done with
//   V_WMMA_F32_16X16X4_F32: D(16ch x 16) += A(16ch x 4pts) * Ones(4x16),
//   accumulated over 4 K-chunks; two chains cover the 32 channels.
// - A operand built directly in the documented VGPR layout
//   (lanes 0-15: K0/K1, lanes 16-31: K2/K3) via ds_bpermute broadcasts.
// - wid forced into an SGPR via readfirstlane so the value base / out base
//   are scalar -> gathers use SADDR + 32-bit byte VOFFSET (1 VALU/address).

typedef __attribute__((ext_vector_type(2))) float v2f;
typedef __attribute__((ext_vector_type(8))) float v8f;

#define BS   8
#define NQ   900
#define NH   8
#define DCH  32
#define NK   22223   // 100*167 + 50*84 + 25*42 + 13*21
#define NWAVES (BS * NQ * NH)   // 57600

__global__ __launch_bounds__(256) void msda_wmma_kernel(
    const float* __restrict__ value,   // (BS, NK, NH, DCH)
    const float* __restrict__ loc,     // (BS, NQ, NH, 4, 4, 2)
    const float* __restrict__ aw,      // (BS, NQ, NH, 4, 4)
    float* __restrict__ out)           // (BS, NQ, NH*DCH)
{
  const int lane = threadIdx.x & 31;
  // wid is wave-uniform; readfirstlane makes that provable -> SGPR addressing.
  const int wid = __builtin_amdgcn_readfirstlane(
      blockIdx.x * 8 + (threadIdx.x >> 5));
  if (wid >= NWAVES) return;           // wave-uniform; EXEC stays all-ones

  const int b   = wid / (NQ * NH);
  const int rem = wid - b * (NQ * NH);
  const int q   = rem >> 3;
  const int h   = rem & 7;

  // ---- meta phase: lane = point (lanes 0..15) ----
  int   gidx[4];   // clamped flat key index per corner
  float gcw[4];    // attention*bilinear weight per corner (0 if OOB)

  if (lane < 16) {
    const int p = lane;
    const int qh = (b * NQ + q) * NH + h;    // scalar
    const float lx = loc[(size_t)qh * 32 + 2 * p];
    const float ly = loc[(size_t)qh * 32 + 2 * p + 1];
    const float w  = aw[(size_t)qh * 16 + p];

    const int l  = p >> 2;
    const int Wi = (l == 0) ? 167 : (l == 1) ? 84 : (l == 2) ? 42 : 21;
    const int Hi = (l == 0) ? 100 : (l == 1) ? 50 : (l == 2) ? 25 : 13;
    const int lb = (l == 0) ? 0 : (l == 1) ? 16700 : (l == 2) ? 20900 : 21950;

    const float x   = lx * (float)Wi - 0.5f;
    const float y   = ly * (float)Hi - 0.5f;
    const float x0f = floorf(x);
    const float y0f = floorf(y);
    const int   ix0 = (int)x0f;
    const int   iy0 = (int)y0f;
    const float fx  = x - x0f, fy = y - y0f;
    const float gx0 = 1.0f - fx, gy0 = 1.0f - fy;

#pragma unroll
    for (int cy = 0; cy < 2; ++cy) {
#pragma unroll
      for (int cx = 0; cx < 2; ++cx) {
        const int c  = cy * 2 + cx;
        const int xi = ix0 + cx;
        const int yi = iy0 + cy;
        const bool valid = (xi >= 0) && (xi < Wi) && (yi >= 0) && (yi < Hi);
        const int xc = min(max(xi, 0), Wi - 1);
        const int yc = min(max(yi, 0), Hi - 1);
        gidx[c] = lb + yc * Wi + xc;
        const float bw = (cx ? fx : gx0) * (cy ? fy : gy0);
        gcw[c] = valid ? (w * bw) : 0.0f;
      }
    }
  }

  // ---- gather + WMMA reduction ----
  // Scalar (SGPR) base; per-lane 32-bit BYTE offset -> SADDR+VOFFSET loads.
  // Per key: 256 floats = 1024 bytes.
  const char* __restrict__ vbq = (const char*)(value
      + (size_t)b * ((size_t)NK * NH * DCH)
      + (size_t)h * DCH);
  const uint32_t chOff = (uint32_t)(lane & 15) << 2;   // channel byte offset
  const int tsel = (lane >> 4) << 1;   // 0 for lanes 0-15, 2 for lanes 16-31

  v8f cLo = {};   // channels 0..15  (row M = channel)
  v8f cHi = {};   // channels 16..31
  const v2f ones = {1.0f, 1.0f};       // B = all-ones: D[m,n] = sum_k A[m,k]

#pragma unroll
  for (int j = 0; j < 4; ++j) {        // K-chunks of 4 points
    v2f aLo;
    v2f aHi;
#pragma unroll
    for (int v = 0; v < 2; ++v) {      // A VGPR 0/1 -> point 4j+v (+tsel in hi half)
      const int src = 4 * j + v + tsel;
      float accLo, accHi;
#pragma unroll
      for (int c = 0; c < 4; ++c) {    // bilinear corners
        const int   gi = __shfl(gidx[c], src);
        const float gw = __shfl(gcw[c], src);
        const uint32_t bo = ((uint32_t)gi << 10) | chOff;  // byte offset
        const float vLo = *(const float*)(vbq + bo);       // channel (lane&15)
        const float vHi = *(const float*)(vbq + bo + 64);  // channel +16
        if (c == 0) {
          accLo = gw * vLo;
          accHi = gw * vHi;
        } else {
          accLo = fmaf(gw, vLo, accLo);
          accHi = fmaf(gw, vHi, accHi);
        }
      }
      aLo[v] = accLo;
      aHi[v] = accHi;
    }
    // D = A x Ones + C  : sums the chunk's 4 points into per-channel rows
    cLo = __builtin_amdgcn_wmma_f32_16x16x4_f32(false, aLo, false, ones,
                                                (short)0, cLo, false, false);
    cHi = __builtin_amdgcn_wmma_f32_16x16x4_f32(false, aHi, false, ones,
                                                (short)0, cHi, false, false);
  }

  // ---- store: C/D layout: VGPR r, lane 0 -> M=r, lane 16 -> M=r+8 ----
  float* op = out + (size_t)(b * NQ + q) * (NH * DCH) + (size_t)h * DCH;
  if ((lane & 15) == 0) {
    const int off = (lane >> 4) * 8;   // 0 (lane 0) or 8 (lane 16)
#pragma unroll
    for (int r = 0; r < 8; ++r) {
      op[off + r]      = cLo[r];       // channels 0..15
      op[off + r + 16] = cHi[r];       // channels 16..31
    }
  }
}

extern "C" void kernel_launch(void* const* d_in, const int* in_sizes, int n_in,
                              void* d_out, int out_size, void* d_ws, size_t ws_size,
                              hipStream_t stream) {
  const float* value = (const float*)d_in[0];
  const float* loc   = (const float*)d_in[1];
  const float* aw    = (const float*)d_in[2];
  float* out         = (float*)d_out;

  dim3 block(256);                  // 8 waves per block
  dim3 grid(NWAVES / 8);            // 7200 blocks, exact
  msda_wmma_kernel<<<grid, block, 0, stream>>>(value, loc, aw, out);
}